// LinearBlendSkinning_6227702579657
// MI455X (gfx1250) — compile-verified
//
#include <hip/hip_runtime.h>

typedef __attribute__((ext_vector_type(2))) float v2f;
typedef __attribute__((ext_vector_type(8))) float v8f;

#define NJ 55
#define KPAD 56            // 14 WMMA steps of K=4
#define WAVES_PER_BLOCK 8
#define VERTS_PER_BLOCK 128
#define W_STRIDE 57        // odd padded stride -> conflict-free A-fragment reads

__global__ __launch_bounds__(256)
void lbs_wmma_f32_kernel(const float* __restrict__ points,
                         const float* __restrict__ weights,
                         const float* __restrict__ se3,
                         float* __restrict__ out,
                         int nverts)
{
    __shared__ float ldsW[VERTS_PER_BLOCK * W_STRIDE]; // 128 x 57 weights tile (padded w/ zeros)
    __shared__ float ldsS[KPAD * 16];                  // SE3 flattened 55x16, row 55 zeroed
    __shared__ float ldsA[WAVES_PER_BLOCK][16 * 17];   // per-wave blended-matrix tile

    const int tid   = threadIdx.x;
    const int lane  = tid & 31;
    const int wave  = tid >> 5;
    const int vblk  = blockIdx.x * VERTS_PER_BLOCK;
    const int rowsValid = min(VERTS_PER_BLOCK, nverts - vblk);

    // --- stage SE3 (55x16 floats, row-major == jkl flattened) into LDS, K-pad row 55 = 0
    for (int i = tid; i < KPAD * 16; i += 256) {
        const int j = i >> 4;
        ldsS[i] = (j < NJ) ? se3[i] : 0.0f;
    }

    // --- stage weights tile into LDS, coalesced reads, zero padding (cols >= 55, rows >= rowsValid)
    for (int i = tid; i < VERTS_PER_BLOCK * W_STRIDE; i += 256) {
        const int r = i / W_STRIDE;
        const int c = i - r * W_STRIDE;
        float v = 0.0f;
        if (c < NJ && r < rowsValid)
            v = weights[(size_t)(vblk + r) * NJ + c];
        ldsW[i] = v;
    }
    __syncthreads();

    // --- 16x16 output tile per wave, K = 56 in steps of 4 via V_WMMA_F32_16X16X4_F32
    const int half = lane >> 4;   // 0: lanes 0-15, 1: lanes 16-31
    const int l16  = lane & 15;
    const float* wRow = &ldsW[(wave * 16 + l16) * W_STRIDE]; // A row m = lane&15

    v8f acc = {};
#pragma unroll
    for (int k0 = 0; k0 < KPAD; k0 += 4) {
        const int ka = k0 + half * 2; // this lane-half's K pair: {ka, ka+1}
        v2f a, b;
        a.x = wRow[ka];               // A[m][ka]
        a.y = wRow[ka + 1];           // A[m][ka+1]
        b.x = ldsS[ka * 16 + l16];        // B[ka][n]
        b.y = ldsS[(ka + 1) * 16 + l16];  // B[ka+1][n]
        acc = __builtin_amdgcn_wmma_f32_16x16x4_f32(
            /*neg_a=*/false, a, /*neg_b=*/false, b,
            /*c_mod=*/(short)0, acc, /*reuse_a=*/false, /*reuse_b=*/false);
    }

    // --- spill D tile (blended 4x4 matrices, 16 vertices) to LDS
    float* aTile = ldsA[wave];
#pragma unroll
    for (int v = 0; v < 8; v++) {
        const int m = v + half * 8;        // D row for VGPR v
        aTile[m * 17 + l16] = acc[v];      // column n = lane&15
    }
    __syncthreads();

    // --- epilogue: lanes 0-15 of each wave apply the blended 3x4 transform to one point
    if (lane < 16) {
        const int vb = wave * 16 + l16;
        if (vb < rowsValid) {
            const size_t n = (size_t)(vblk + vb);
            const float px = points[n * 3 + 0];
            const float py = points[n * 3 + 1];
            const float pz = points[n * 3 + 2];
            const float* A = &aTile[l16 * 17]; // A[m][c], c = 4*k + l
#pragma unroll
            for (int k = 0; k < 3; k++) {
                const float r = A[4 * k + 0] * px + A[4 * k + 1] * py +
                                A[4 * k + 2] * pz + A[4 * k + 3];
                out[n * 3 + k] = r;
            }
        }
    }
}

extern "C" void kernel_launch(void* const* d_in, const int* in_sizes, int n_in,
                              void* d_out, int out_size, void* d_ws, size_t ws_size,
                              hipStream_t stream) {
    const float* points  = (const float*)d_in[0]; // (N,3)
    const float* weights = (const float*)d_in[1]; // (N,55)
    const float* se3     = (const float*)d_in[2]; // (55,4,4)
    float* out = (float*)d_out;                   // (N,3)

    const int nverts = in_sizes[0] / 3;
    const int blocks = (nverts + VERTS_PER_BLOCK - 1) / VERTS_PER_BLOCK;
    hipLaunchKernelGGL(lbs_wmma_f32_kernel, dim3(blocks), dim3(256), 0, stream,
                       points, weights, se3, out, nverts);
}